// AutoAttention_Layer_7954279432666
// MI455X (gfx1250) — compile-verified
//
#include <hip/hip_runtime.h>
#include <hip/hip_bf16.h>

typedef __attribute__((ext_vector_type(2))) float v2f;
typedef __attribute__((ext_vector_type(8))) float v8f;
typedef __attribute__((ext_vector_type(4))) unsigned int v4u;
typedef __attribute__((ext_vector_type(8))) int v8i;
typedef __attribute__((ext_vector_type(4))) int v4i;

#define NEG_PAD (-4294967296.0f)   // float32(-2^32+1) rounds to -2^32
#define KSTR 68                    // 200x64 k tile padded: 68 floats = 17*16B rows

#if defined(__gfx1250__) && __has_builtin(__builtin_amdgcn_tensor_load_to_lds)
#define USE_TDM 1
#else
#define USE_TDM 0
#endif

// ---------------------------------------------------------------------------
// Kernel 1: Mmat = Wq @ Wk^T  (64x64 = 4x4 tiles of 16x16, one wave per tile)
// ---------------------------------------------------------------------------
__global__ __launch_bounds__(512) void k_prep_mmat(const float* __restrict__ Wq,
                                                   const float* __restrict__ Wk,
                                                   float* __restrict__ Mmat) {
    int tid  = threadIdx.x;
    int wave = tid >> 5, lane = tid & 31;
    int half = lane >> 4, r = lane & 15;
    int mi = wave >> 2, ni = wave & 3;

    v8f c = {};
    for (int kk = 0; kk < 16; ++kk) {
        int k0 = kk * 4 + 2 * half;
        v2f a, b;
        // A = Wq, row-major: A[m][k]
        a.x = Wq[(mi * 16 + r) * 64 + k0];
        a.y = Wq[(mi * 16 + r) * 64 + k0 + 1];
        // B = Wk^T: B[k][n] = Wk[n][k]
        b.x = Wk[(ni * 16 + r) * 64 + k0];
        b.y = Wk[(ni * 16 + r) * 64 + k0 + 1];
        c = __builtin_amdgcn_wmma_f32_16x16x4_f32(false, a, false, b,
                                                  (short)0, c, false, false);
    }
    for (int i = 0; i < 8; ++i) {
        int row = mi * 16 + i + 8 * half;
        int col = ni * 16 + r;
        Mmat[row * 64 + col] = c[i];
    }
}

// ---------------------------------------------------------------------------
// Kernel 2: qbar[b] = sum_f fs[f]*q[b,f,:]  then  qtil = qbar @ Mmat  (WMMA)
// 16 batches per block, 4 waves (one N-tile each)
// ---------------------------------------------------------------------------
__global__ __launch_bounds__(128) void k_qtil(const float* __restrict__ q,
                                              const float* __restrict__ fs,
                                              const float* __restrict__ Mmat,
                                              float* __restrict__ qtil) {
    __shared__ float qbar[16][65];   // pad: A-frag reads hit distinct banks
    __shared__ float fsl[64];
    int tid = threadIdx.x;
    int b0  = blockIdx.x * 16;

    if (tid < 64) fsl[tid] = fs[tid];
    __syncthreads();

    int d = tid & 63;
    for (int bb = tid >> 6; bb < 16; bb += 2) {
        const float* qb = q + (size_t)(b0 + bb) * 64 * 64;
        float s = 0.0f;
        for (int f = 0; f < 64; ++f) s += fsl[f] * qb[f * 64 + d];  // coalesced in d
        qbar[bb][d] = s;
    }
    __syncthreads();

    int wave = tid >> 5, lane = tid & 31;
    int half = lane >> 4, r = lane & 15;
    int ni = wave;

    v8f c = {};
    for (int kk = 0; kk < 16; ++kk) {
        int k0 = kk * 4 + 2 * half;
        v2f a, b;
        a.x = qbar[r][k0];
        a.y = qbar[r][k0 + 1];
        b.x = Mmat[k0 * 64 + ni * 16 + r];
        b.y = Mmat[(k0 + 1) * 64 + ni * 16 + r];
        c = __builtin_amdgcn_wmma_f32_16x16x4_f32(false, a, false, b,
                                                  (short)0, c, false, false);
    }
    for (int i = 0; i < 8; ++i)
        qtil[(size_t)(b0 + i + 8 * half) * 64 + ni * 16 + r] = c[i];
}

// ---------------------------------------------------------------------------
// Kernel 3: per-batch  score -> sigmoid/mask -> softmax -> abar
// k_b (200x64) staged once in LDS via TDM (hardware row-padding to KSTR=68),
// so k is read exactly once from HBM with a single DMA per workgroup.
// ---------------------------------------------------------------------------
__global__ __launch_bounds__(256) void k_attn(const float* __restrict__ kmat,
                                              const int* __restrict__ kes_len,
                                              const float* __restrict__ biasp,
                                              const float* __restrict__ qtil,
                                              float* __restrict__ abar_out) {
    __shared__ __align__(16) float klds[200 * KSTR];
    __shared__ float sc[256];
    __shared__ float red[256];
    __shared__ float qt[64];

    int tid = threadIdx.x;
    int b   = blockIdx.x;

#if USE_TDM
    if (tid < 32) {   // wave 0 issues one TDM DMA for the whole tile
        unsigned lds_base = (unsigned)(uintptr_t)&klds[0];   // low 32 bits = LDS offset
        unsigned long long gaddr =
            (unsigned long long)(uintptr_t)(kmat + (size_t)b * 200 * 64);
        // D# group0: count=1 | lds_addr | global_addr[56:0] | type=2
        v4u g0;
        g0.x = 1u;
        g0.y = lds_base;
        g0.z = (unsigned)(gaddr & 0xFFFFFFFFull);
        g0.w = (unsigned)((gaddr >> 32) & 0x01FFFFFFull) | (2u << 30);
        // D# group1: data_size=2 (4B), pad_enable=1,
        // pad_interval=5 (pad every 64 DWORDs), pad_amount=3 (4 DWORDs) -> stride 68
        v8i g1;
        g1[0] = (2 << 16) | (1 << 20) | (5 << 22) | (3 << 25);
        g1[1] = (64  << 16);   // atomic_barrier_addr=0 | tensor_dim0[15:0]=64
        g1[2] = (200 << 16);   // tensor_dim0[31:16]=0  | tensor_dim1[15:0]=200
        g1[3] = (64  << 16);   // tensor_dim1[31:16]=0  | tile_dim0=64
        g1[4] = 200;           // tile_dim1=200 | tile_dim2=0
        g1[5] = 64;            // tensor_dim0_stride[31:0]=64
        g1[6] = 0;             // stride0[47:32]=0 | tensor_dim1_stride lo (unused, 2D)
        g1[7] = 0;
        v4i gz4 = {0, 0, 0, 0};           // groups 2/3: tensor is 2-D
        v8i gz8 = {0, 0, 0, 0, 0, 0, 0, 0};
        __builtin_amdgcn_tensor_load_to_lds(g0, g1, gz4, gz4, gz8, 0);
        __builtin_amdgcn_s_wait_tensorcnt(0);
    }
#else
    // fallback: stage k_b with float4s; 16B-aligned stores (KSTR*4 = 272 = 17*16)
    const float4* kb = (const float4*)(kmat + (size_t)b * 200 * 64);
    for (int i = tid; i < 200 * 16; i += 256) {
        int row = i >> 4, c4 = i & 15;
        float4 val = kb[i];
        *(float4*)&klds[row * KSTR + c4 * 4] = val;
    }
#endif
    if (tid < 64) qt[tid] = qtil[(size_t)b * 64 + tid];
    __syncthreads();

    int   len    = kes_len[b];
    float bias64 = biasp[0] * 64.0f;

    // logits
    float logit = -__builtin_inff();
    if (tid < 200) {
        float s = 0.0f;
        for (int d = 0; d < 64; ++d) s += qt[d] * klds[tid * KSTR + d];
        s += bias64;
        float sig = 1.0f / (1.0f + expf(-s));
        float m   = (tid < len) ? sig : NEG_PAD;
        logit = m * 0.125f;                 // 1/sqrt(64)
    }
    sc[tid] = logit;

    // max-reduce
    red[tid] = logit;
    __syncthreads();
    for (int s = 128; s > 0; s >>= 1) {
        if (tid < s) red[tid] = fmaxf(red[tid], red[tid + s]);
        __syncthreads();
    }
    float mx = red[0];
    __syncthreads();

    // sum-reduce of exp
    float e = (tid < 200) ? expf(sc[tid] - mx) : 0.0f;
    red[tid] = e;
    __syncthreads();
    for (int s = 128; s > 0; s >>= 1) {
        if (tid < s) red[tid] += red[tid + s];
        __syncthreads();
    }
    float inv = 1.0f / red[0];
    __syncthreads();
    if (tid < 200) sc[tid] = e * inv;       // probabilities a_t
    __syncthreads();

    // abar[d] = sum_t a_t * k[t][d]   (4 partial sums over t, then combine)
    int d = tid & 63, part = tid >> 6;
    float s = 0.0f;
    for (int t = part; t < 200; t += 4) s += sc[t] * klds[t * KSTR + d];
    red[tid] = s;
    __syncthreads();
    if (tid < 64)
        abar_out[(size_t)b * 64 + tid] =
            red[tid] + red[tid + 64] + red[tid + 128] + red[tid + 192];
}

// ---------------------------------------------------------------------------
// Kernel 4: out = abar @ Wv   (16 batches/block, 4 waves, WMMA)
// ---------------------------------------------------------------------------
__global__ __launch_bounds__(128) void k_out(const float* __restrict__ abar,
                                             const float* __restrict__ Wv,
                                             float* __restrict__ out) {
    __shared__ float at[16][65];
    int tid = threadIdx.x;
    int b0  = blockIdx.x * 16;

    for (int i = tid; i < 16 * 64; i += 128)
        at[i >> 6][i & 63] = abar[(size_t)b0 * 64 + i];
    __syncthreads();

    int wave = tid >> 5, lane = tid & 31;
    int half = lane >> 4, r = lane & 15;
    int ni = wave;

    v8f c = {};
    for (int kk = 0; kk < 16; ++kk) {
        int k0 = kk * 4 + 2 * half;
        v2f a, b;
        a.x = at[r][k0];
        a.y = at[r][k0 + 1];
        b.x = Wv[k0 * 64 + ni * 16 + r];
        b.y = Wv[(k0 + 1) * 64 + ni * 16 + r];
        c = __builtin_amdgcn_wmma_f32_16x16x4_f32(false, a, false, b,
                                                  (short)0, c, false, false);
    }
    for (int i = 0; i < 8; ++i)
        out[(size_t)(b0 + i + 8 * half) * 64 + ni * 16 + r] = c[i];
}

// ---------------------------------------------------------------------------
extern "C" void kernel_launch(void* const* d_in, const int* in_sizes, int n_in,
                              void* d_out, int out_size, void* d_ws, size_t ws_size,
                              hipStream_t stream) {
    const float* q    = (const float*)d_in[0];   // (1024, 64, 64)
    const float* kmat = (const float*)d_in[1];   // (1024, 200, 64)
    // d_in[2] = v : provably unused by the reference
    const int*   len  = (const int*)  d_in[3];   // (1024, 1)
    const float* fs   = (const float*)d_in[4];   // (64, 1)
    const float* bias = (const float*)d_in[5];   // (1,)
    const float* Wq   = (const float*)d_in[6];   // (64, 64)
    const float* Wk   = (const float*)d_in[7];   // (64, 64)
    const float* Wv   = (const float*)d_in[8];   // (64, 64)
    float*       out  = (float*)d_out;           // (1024, 1, 64)

    float* Mmat = (float*)d_ws;                  //   4096 floats
    float* qtil = Mmat + 64 * 64;                //  65536 floats
    float* abar = qtil + 1024 * 64;              //  65536 floats

    k_prep_mmat<<<1,    512, 0, stream>>>(Wq, Wk, Mmat);
    k_qtil     <<<64,   128, 0, stream>>>(q, fs, Mmat, qtil);
    k_attn     <<<1024, 256, 0, stream>>>(kmat, len, bias, qtil, abar);
    k_out      <<<64,   128, 0, stream>>>(abar, Wv, out);
}